// MPN_75411035783819
// MI455X (gfx1250) — compile-verified
//
#include <hip/hip_runtime.h>

#define NN 100000
#define NE 1600000
#define D  64
#define WSTR 132   // padded LDS row stride (floats): bank = (4n+k)&63 -> conflict-free WMMA feeds

typedef __attribute__((ext_vector_type(2))) float v2f;
typedef __attribute__((ext_vector_type(8))) float v8f;

// Guaranteed-hardware fp32 atomic add (no CAS fallback). Fire-and-forget;
// S_ENDPGM's implicit wait-idle covers completion, stream order covers layers.
__device__ __forceinline__ void hw_atomic_add_f32(float* p, float v) {
    asm volatile("global_atomic_add_f32 %0, %1, off" :: "v"(p), "v"(v) : "memory");
}

// ---------------------------------------------------------------------------
// Scatter-add aggregation: agg[dst[e]] += x[src[e]].
// 16 lanes per edge, float4 per lane -> one coalesced 256B row gather per
// edge. x (25.6MB) and agg both fit in the 192MB L2, so this is bound by
// L2 bandwidth / L2 atomic throughput, which native fp32 atomics maximize.
// ---------------------------------------------------------------------------
__global__ __launch_bounds__(256) void gnn_scatter_kernel(
    const float* __restrict__ x, const int* __restrict__ src,
    const int* __restrict__ dst, float* __restrict__ agg)
{
    unsigned t = blockIdx.x * 256u + threadIdx.x;
    unsigned e = t >> 4;
    if (e >= NE) return;
    unsigned f = (t & 15u) * 4u;
    int s = src[e];
    int d = dst[e];
    const float4 v = *reinterpret_cast<const float4*>(x + (size_t)s * D + f);
    float* o = agg + (size_t)d * D + f;
    hw_atomic_add_f32(o + 0, v.x);
    hw_atomic_add_f32(o + 1, v.y);
    hw_atomic_add_f32(o + 2, v.z);
    hw_atomic_add_f32(o + 3, v.w);
}

// ---------------------------------------------------------------------------
// Fused dense layer: out = relu?( [agg|x] @ [w_rel; w_root] + b )
// Single K=128 GEMM via V_WMMA_F32_16X16X4_F32 (fp32-exact vs reference).
// Block = 256 threads = 8 waves; wave -> one 16x16 tile (32 rows x 64 cols
// per block). Both operands staged in padded LDS: inner loop is pure
// ds_load_b64 + v_wmma, conflict-free.
// ---------------------------------------------------------------------------
template <bool RELU>
__global__ __launch_bounds__(256) void gnn_dense_kernel(
    const float* __restrict__ agg, const float* __restrict__ xin,
    const float* __restrict__ w_rel, const float* __restrict__ w_root,
    const float* __restrict__ bias, float* __restrict__ out)
{
    __shared__ float Wt[D * WSTR];   // Wt[n*WSTR + k]  (weights, transposed)
    __shared__ float Xs[32 * WSTR];  // Xs[r*WSTR + k]  (32-row [agg|x] tile)

    const int rowBlock = blockIdx.x * 32;

    // Stage transposed weights: n-fast => coalesced 256B reads of w_* rows.
    for (int idx = threadIdx.x; idx < D * 128; idx += 256) {
        int n = idx & 63;
        int k = idx >> 6;
        Wt[n * WSTR + k] = (k < D) ? w_rel[k * D + n] : w_root[(k - D) * D + n];
    }
    // Stage the 32x128 activation tile: float4-coalesced.
    for (int q = threadIdx.x; q < 32 * 32; q += 256) {
        int row = q >> 5;
        int qc  = (q & 31) * 4;
        int grow = rowBlock + row;
        if (grow >= NN) grow = NN - 1;           // clamp: keeps EXEC full later
        const float* sp = (qc < D) ? (agg + (size_t)grow * D + qc)
                                   : (xin + (size_t)grow * D + (qc - D));
        *reinterpret_cast<float4*>(&Xs[row * WSTR + qc]) =
            *reinterpret_cast<const float4*>(sp);
    }
    __syncthreads();

    const int wave    = threadIdx.x >> 5;
    const int lane    = threadIdx.x & 31;
    const int rowTile = wave >> 2;          // 0..1
    const int colTile = wave & 3;           // 0..3
    const int colBase = colTile * 16;
    const int mn      = lane & 15;          // M for A-frag, N for B-frag
    const int khalf   = (lane >> 4) << 1;   // 0 or 2

    const float* arow = &Xs[(rowTile * 16 + mn) * WSTR];
    const float* brow = &Wt[(colBase + mn) * WSTR];

    v8f c = {};
#pragma unroll
    for (int kk = 0; kk < 128; kk += 4) {
        const int k = kk + khalf;
        v2f a = *reinterpret_cast<const v2f*>(arow + k);   // ds_load_b64
        v2f b = *reinterpret_cast<const v2f*>(brow + k);   // ds_load_b64
        // (neg_a, A, neg_b, B, c_mod, C, reuse_a, reuse_b)
        c = __builtin_amdgcn_wmma_f32_16x16x4_f32(
                false, a, false, b, (short)0, c, false, false);
    }

    const float bb      = bias[colBase + mn];
    const int   rowHalf = rowBlock + rowTile * 16 + ((lane >> 4) * 8);
#pragma unroll
    for (int r = 0; r < 8; r++) {
        int   row = rowHalf + r;
        float v   = c[r] + bb;
        if (RELU) v = fmaxf(v, 0.0f);
        if (row < NN) out[(size_t)row * D + (colBase + mn)] = v;
    }
}

// ---------------------------------------------------------------------------
extern "C" void kernel_launch(void* const* d_in, const int* in_sizes, int n_in,
                              void* d_out, int out_size, void* d_ws, size_t ws_size,
                              hipStream_t stream)
{
    (void)in_sizes; (void)n_in; (void)out_size; (void)ws_size;

    const float* x      = (const float*)d_in[0];
    const int*   ei     = (const int*)d_in[1];
    const int*   src    = ei;
    const int*   dst    = ei + NE;
    const float* w_rel1 = (const float*)d_in[2];
    const float* w_root1= (const float*)d_in[3];
    const float* b1     = (const float*)d_in[4];
    const float* w_rel2 = (const float*)d_in[5];
    const float* w_root2= (const float*)d_in[6];
    const float* b2     = (const float*)d_in[7];
    const float* w_rel3 = (const float*)d_in[8];
    const float* w_root3= (const float*)d_in[9];
    const float* b3     = (const float*)d_in[10];
    float*       out    = (float*)d_out;

    const size_t featElems = (size_t)NN * D;
    float* agg = (float*)d_ws;
    float* h1  = agg + featElems;
    float* h2  = h1 + featElems;
    const size_t aggBytes = featElems * sizeof(float);

    const dim3 blk(256);
    const dim3 grdS((NE * 16 + 255) / 256);   // 100000 blocks
    const dim3 grdD((NN + 31) / 32);          // 3125 blocks

    // Layer 1
    hipMemsetAsync(agg, 0, aggBytes, stream);
    gnn_scatter_kernel<<<grdS, blk, 0, stream>>>(x, src, dst, agg);
    gnn_dense_kernel<true><<<grdD, blk, 0, stream>>>(agg, x, w_rel1, w_root1, b1, h1);

    // Layer 2
    hipMemsetAsync(agg, 0, aggBytes, stream);
    gnn_scatter_kernel<<<grdS, blk, 0, stream>>>(h1, src, dst, agg);
    gnn_dense_kernel<true><<<grdD, blk, 0, stream>>>(agg, h1, w_rel2, w_root2, b2, h2);

    // Layer 3
    hipMemsetAsync(agg, 0, aggBytes, stream);
    gnn_scatter_kernel<<<grdS, blk, 0, stream>>>(h2, src, dst, agg);
    gnn_dense_kernel<false><<<grdD, blk, 0, stream>>>(agg, h2, w_rel3, w_root3, b3, out);
}